// AutoCorrelation_18511309046318
// MI455X (gfx1250) — compile-verified
//
#include <hip/hip_runtime.h>
#include <math.h>

// ---------------------------------------------------------------------------
// AutoCorrelation via radix-16 matmul-FFT on CDNA5 WMMA (gfx1250, wave32).
// 2048 lanes (B=16 x D=128), S=4096 = 16^3 -> 3 DFT16 stages per transform.
// Each DFT16 stage tile is a 16x16 complex matmul done with 16 chained
// V_WMMA_F32_16X16X4_F32 ops (exact fp32).
// ---------------------------------------------------------------------------

#define TWO_PI 6.28318530717958647692f

typedef float v2f __attribute__((ext_vector_type(2)));
typedef float v8f __attribute__((ext_vector_type(8)));

__device__ __forceinline__ v8f wmma4(v2f a, v2f b, v8f c) {
  // D = A(16x4) * B(4x16) + C(16x16), fp32
  return __builtin_amdgcn_wmma_f32_16x16x4_f32(false, a, false, b, (short)0, c,
                                               false, false);
}

// Stage addressing for N = 4096 = 16*16*16 DIF network.
// stage0: addr = g + 256*j           (g in [0,256), twiddle w_4096^(g*k))
// stage1: addr = 256*(g>>4)+(g&15)+16*j  (twiddle w_256^((g&15)*k))
// stage2: addr = 16*g + j            (no twiddle)
__device__ __forceinline__ int stage_base(int s, int g) {
  return (s == 0) ? g : ((s == 1) ? ((((g >> 4) << 8)) | (g & 15)) : (g << 4));
}
__device__ __forceinline__ int stage_stride(int s) {
  return (s == 0) ? 256 : ((s == 1) ? 16 : 1);
}
__device__ __forceinline__ int stage_twm(int s, int g) {
  return (s == 0) ? g : ((s == 1) ? (g & 15) : 0);
}
__device__ __forceinline__ int stage_twN(int s) {
  return (s == 0) ? 4096 : ((s == 1) ? 256 : 1);
}

// One 16-group butterfly tile: B = X[j][group] (16x16 complex) gathered from
// LDS into WMMA B-layout, multiplied by the DFT16 matrix held in A-layout
// registers (Cm = cos, Sm = +sin, Sn = -sin), twiddled, written back in place.
template <bool INV>
__device__ __forceinline__ void process_tile(float* __restrict__ re,
                                             float* __restrict__ im, int s,
                                             int gbase, float scale,
                                             const v2f Cm[4], const v2f Sm[4],
                                             const v2f Sn[4]) {
  const int lane = threadIdx.x & 31;
  const int n = lane & 15;   // column (group within tile)
  const int h = lane >> 4;   // lane-half selects K rows
  const int g = gbase + n;
  const int ab = stage_base(s, g);
  const int sj = stage_stride(s);
  const int twm = stage_twm(s, g);
  const int twN = stage_twN(s);

  v2f Bre[4], Bim[4];
#pragma unroll
  for (int kk = 0; kk < 4; ++kk) {
    const int j0 = 4 * kk + 2 * h;  // B rows held by this lane for chunk kk
    const int a0 = ab + sj * j0;
    const int a1 = a0 + sj;
    float xr0 = re[a0], xi0 = im[a0];
    float xr1 = re[a1], xi1 = im[a1];
    if (INV && s != 2) {
      // inverse network: conjugate twiddle applied BEFORE the IDFT16
      const float w = TWO_PI / (float)twN;
      float s0, c0, s1, c1, tr;
      __sincosf((float)((twm * j0) & (twN - 1)) * w, &s0, &c0);
      __sincosf((float)((twm * (j0 + 1)) & (twN - 1)) * w, &s1, &c1);
      tr = xr0 * c0 - xi0 * s0; xi0 = xr0 * s0 + xi0 * c0; xr0 = tr;
      tr = xr1 * c1 - xi1 * s1; xi1 = xr1 * s1 + xi1 * c1; xr1 = tr;
    }
    Bre[kk].x = xr0; Bre[kk].y = xr1;
    Bim[kk].x = xi0; Bim[kk].y = xi1;
  }

  v8f Dre = {};
  v8f Dim = {};
#pragma unroll
  for (int kk = 0; kk < 4; ++kk) {
    if (!INV) {
      // W = C - iS:  Yre = C*Xre + S*Xim ; Yim = C*Xim - S*Xre
      Dre = wmma4(Cm[kk], Bre[kk], Dre);
      Dre = wmma4(Sm[kk], Bim[kk], Dre);
      Dim = wmma4(Cm[kk], Bim[kk], Dim);
      Dim = wmma4(Sn[kk], Bre[kk], Dim);
    } else {
      // W = C + iS:  Yre = C*Xre - S*Xim ; Yim = C*Xim + S*Xre
      Dre = wmma4(Cm[kk], Bre[kk], Dre);
      Dre = wmma4(Sn[kk], Bim[kk], Dre);
      Dim = wmma4(Cm[kk], Bim[kk], Dim);
      Dim = wmma4(Sm[kk], Bre[kk], Dim);
    }
  }

#pragma unroll
  for (int r = 0; r < 8; ++r) {
    const int k = r + 8 * h;          // D-layout: reg r holds row r / r+8
    const int addr = ab + sj * k;
    float yr = Dre[r], yi = Dim[r];
    if (!INV && s != 2) {
      // forward network: twiddle w_twN^{-twm*k} applied AFTER the DFT16
      const float w = -TWO_PI / (float)twN;
      float ss, cc;
      __sincosf((float)((twm * k) & (twN - 1)) * w, &ss, &cc);
      float tr = yr * cc - yi * ss;
      yi = yr * ss + yi * cc;
      yr = tr;
    }
    re[addr] = yr * scale;
    im[addr] = yi * scale;
  }
}

template <bool INV>
__device__ __forceinline__ void fft3(float* re, float* im, const v2f Cm[4],
                                     const v2f Sm[4], const v2f Sn[4]) {
  const int wv = threadIdx.x >> 5;  // 8 waves, 2 tiles each per stage
#pragma unroll
  for (int si = 0; si < 3; ++si) {
    const int s = INV ? (2 - si) : si;
    const float scale = (INV && s == 0) ? (1.0f / 4096.0f) : 1.0f;
    process_tile<INV>(re, im, s, (2 * wv) * 16, scale, Cm, Sm, Sn);
    process_tile<INV>(re, im, s, (2 * wv + 1) * 16, scale, Cm, Sm, Sn);
    __syncthreads();
  }
}

// One workgroup per (b,d) lane. fast=1: lane-contiguous [B,D,S] scratch
// (stride 1); fast=0: strided access straight into [B,S,D] tensors.
__global__ __launch_bounds__(256) void autocorr_fft_wmma_kernel(
    const float* __restrict__ Qp, const float* __restrict__ Kp,
    const float* __restrict__ Vp, float* __restrict__ Op, int fast) {
  __shared__ float xr[4096];
  __shared__ float xi[4096];
  __shared__ float sr[4096];
  __shared__ float si_[4096];

  const int laneid = blockIdx.x;        // 0..2047
  const int b = laneid >> 7;
  const int d = laneid & 127;
  const int tid = threadIdx.x;
  int base, str;
  if (fast) { base = laneid << 12; str = 1; }
  else      { base = b * 524288 + d; str = 128; }

  // DFT16 matrices in WMMA A-layout registers
  v2f Cm[4], Sm[4], Sn[4];
  {
    const int lane = tid & 31;
    const int n = lane & 15;
    const int h = lane >> 4;
#pragma unroll
    for (int kk = 0; kk < 4; ++kk) {
      const int k0 = 4 * kk + 2 * h;
      const float a0 = (float)((n * k0) & 15) * (TWO_PI / 16.0f);
      const float a1 = (float)((n * (k0 + 1)) & 15) * (TWO_PI / 16.0f);
      Cm[kk].x = cosf(a0); Cm[kk].y = cosf(a1);
      Sm[kk].x = sinf(a0); Sm[kk].y = sinf(a1);
      Sn[kk].x = -Sm[kk].x; Sn[kk].y = -Sm[kk].y;
    }
  }

  // ---- FFT(K) ----
  for (int j = 0; j < 16; ++j) {
    const int t = j * 256 + tid;
    xr[t] = Kp[base + t * str];
    xi[t] = 0.0f;
  }
  __syncthreads();
  fft3<false>(xr, xi, Cm, Sm, Sn);
  for (int j = 0; j < 16; ++j) {
    const int t = j * 256 + tid;
    sr[t] = xr[t];
    si_[t] = xi[t];
  }
  // ---- FFT(Q) ----
  for (int j = 0; j < 16; ++j) {
    const int t = j * 256 + tid;
    xr[t] = Qp[base + t * str];
    xi[t] = 0.0f;
  }
  __syncthreads();
  fft3<false>(xr, xi, Cm, Sm, Sn);

  // ---- pointwise FQ * conj(FK) (order-independent) ----
  for (int j = 0; j < 16; ++j) {
    const int t = j * 256 + tid;
    const float a = xr[t], bq = xi[t], c = sr[t], dq = si_[t];
    xr[t] = a * c + bq * dq;
    xi[t] = bq * c - a * dq;
  }
  __syncthreads();

  // ---- inverse FFT -> corr (natural order, scaled by 1/N) ----
  fft3<true>(xr, xi, Cm, Sm, Sn);

  // ---- top-8 (value, lag) by 8 rounds of blockwide argmax ----
  float* w8 = si_;                    // spectrum LDS is free now
  int* d8 = (int*)(si_ + 8);
  float* rv = si_ + 16;
  int* ri = (int*)(si_ + 272);
  for (int k = 0; k < 8; ++k) {
    float bv = -3.4e38f;
    int bi = 0;
    for (int j = 0; j < 16; ++j) {
      const int t = tid * 16 + j;
      const float v = xr[t];
      if (v > bv) { bv = v; bi = t; }
    }
    rv[tid] = bv;
    ri[tid] = bi;
    __syncthreads();
    for (int off = 128; off > 0; off >>= 1) {
      if (tid < off) {
        if (rv[tid + off] > rv[tid]) {
          rv[tid] = rv[tid + off];
          ri[tid] = ri[tid + off];
        }
      }
      __syncthreads();
    }
    if (tid == 0) {
      w8[k] = rv[0];
      d8[k] = ri[0];
      xr[ri[0]] = -3.4e38f;
    }
    __syncthreads();
  }

  // ---- time-delay aggregation over V ----
  for (int j = 0; j < 16; ++j) {
    const int t = j * 256 + tid;
    sr[t] = Vp[base + t * str];
  }
  __syncthreads();
  float wl[8];
  int dl[8];
#pragma unroll
  for (int k = 0; k < 8; ++k) { wl[k] = w8[k]; dl[k] = d8[k]; }
  for (int j = 0; j < 16; ++j) {
    const int t = j * 256 + tid;
    float acc = 0.0f;
#pragma unroll
    for (int k = 0; k < 8; ++k) acc += wl[k] * sr[(t + dl[k]) & 4095];
    Op[base + t * str] = acc;
  }
}

// Batched [z][R][C] -> [z][C][R] tiled transpose (coalesced both directions).
__global__ __launch_bounds__(256) void transpose_rc_kernel(
    const float* __restrict__ in, float* __restrict__ out, int R, int C) {
  __shared__ float tile[32][33];
  const int zoff = blockIdx.z * R * C;
  const int r0 = blockIdx.y * 32;
  const int c0 = blockIdx.x * 32;
  for (int i = threadIdx.y; i < 32; i += 8) {
    tile[i][threadIdx.x] = in[zoff + (r0 + i) * C + (c0 + threadIdx.x)];
  }
  __syncthreads();
  for (int i = threadIdx.y; i < 32; i += 8) {
    out[zoff + (c0 + i) * R + (r0 + threadIdx.x)] = tile[threadIdx.x][i];
  }
}

extern "C" void kernel_launch(void* const* d_in, const int* in_sizes, int n_in,
                              void* d_out, int out_size, void* d_ws,
                              size_t ws_size, hipStream_t stream) {
  (void)in_sizes; (void)n_in; (void)out_size;
  const float* Q = (const float*)d_in[0];
  const float* K = (const float*)d_in[1];
  const float* V = (const float*)d_in[2];
  float* out = (float*)d_out;

  const size_t M = (size_t)16 * 128 * 4096;           // elems per tensor
  const bool fast = (d_ws != nullptr) && (ws_size >= 4 * M * sizeof(float));

  if (fast) {
    float* ws = (float*)d_ws;
    float* qT = ws;
    float* kT = ws + M;
    float* vT = ws + 2 * M;
    float* oT = ws + 3 * M;
    dim3 tb(32, 8);
    // [B,S,D] -> [B,D,S]  (R=4096, C=128)
    transpose_rc_kernel<<<dim3(4, 128, 16), tb, 0, stream>>>(Q, qT, 4096, 128);
    transpose_rc_kernel<<<dim3(4, 128, 16), tb, 0, stream>>>(K, kT, 4096, 128);
    transpose_rc_kernel<<<dim3(4, 128, 16), tb, 0, stream>>>(V, vT, 4096, 128);
    autocorr_fft_wmma_kernel<<<2048, 256, 0, stream>>>(qT, kT, vT, oT, 1);
    // [B,D,S] -> [B,S,D]  (R=128, C=4096)
    transpose_rc_kernel<<<dim3(128, 4, 16), tb, 0, stream>>>(oT, out, 128, 4096);
  } else {
    autocorr_fft_wmma_kernel<<<2048, 256, 0, stream>>>(Q, K, V, out, 0);
  }
}